// StatMechSystem_36146444763879
// MI455X (gfx1250) — compile-verified
//
#include <hip/hip_runtime.h>

typedef __attribute__((ext_vector_type(2))) float v2f;
typedef __attribute__((ext_vector_type(8))) float v8f;

#define NX    256
#define NP    256
#define UNITS 256

// ---------------------------------------------------------------------------
// Kernel 1: Th[p] = prod_u sum_t (xp@W)[p,u,t] * xp[p,(u-1)%UNITS,t]
// 256 outputs, one thread each — negligible cost, runs before the main kernel.
// ---------------------------------------------------------------------------
__global__ void statmech_th_kernel(const float* __restrict__ xp,
                                   const float* __restrict__ W,
                                   float* __restrict__ Th) {
    int p = blockIdx.x * blockDim.x + threadIdx.x;
    if (p >= NP) return;
    const float W00 = W[0], W01 = W[1], W10 = W[2], W11 = W[3]; // W[s][t]
    const float* row = xp + (size_t)p * UNITS * 2;
    float prev0 = row[(UNITS - 1) * 2 + 0];
    float prev1 = row[(UNITS - 1) * 2 + 1];
    float th = 1.0f;
    for (int u = 0; u < UNITS; ++u) {
        float c0 = row[u * 2 + 0];
        float c1 = row[u * 2 + 1];
        float w0 = c0 * W00 + c1 * W10;   // xpW[p,u,0]
        float w1 = c0 * W01 + c1 * W11;   // xpW[p,u,1]
        th *= (w0 * prev0 + w1 * prev1);  // dot with rolled xp (u-1)
        prev0 = c0; prev1 = c1;
    }
    Th[p] = th;
}

// ---------------------------------------------------------------------------
// Kernel 2: out[b,p] = Th[p] * prod_u s[b,p,u],
//           s[b,p,u] = sum_t (x@W)[b,u,t] * xp[p,u,t]
// One wave32 per 16x16 output tile. Per unit u, one V_WMMA_F32_16X16X4_F32
// computes the rank-2 s-tile (K=0,1 live in lanes 0-15; lanes 16-31 feed
// K=2,3 with zeros). The 8-VGPR f32 D-tile is folded into a running product.
// ---------------------------------------------------------------------------
__global__ void statmech_tv_kernel(const float* __restrict__ x,
                                   const float* __restrict__ xp,
                                   const float* __restrict__ W,
                                   const float* __restrict__ Th,
                                   float* __restrict__ out) {
    const int lane = threadIdx.x & 31;
    const int wid  = (blockIdx.x * blockDim.x + threadIdx.x) >> 5; // 0..255
    const int bbase = (wid >> 4) * 16;
    const int pbase = (wid & 15) * 16;
    const int l15   = lane & 15;
    const float m   = (lane < 16) ? 1.0f : 0.0f; // zero K=2,3 contributions

    // Pre-scale W by m so A-matrix hi lanes are zero for free.
    const float W00 = W[0] * m, W01 = W[1] * m;
    const float W10 = W[2] * m, W11 = W[3] * m;

    // Lane l15 owns row b = bbase+l15 of x and column p = pbase+l15 of xp.
    const float4* xrow = (const float4*)(x  + (size_t)(bbase + l15) * UNITS * 2);
    const float4* prow = (const float4*)(xp + (size_t)(pbase + l15) * UNITS * 2);

    v8f acc;
#pragma unroll
    for (int i = 0; i < 8; ++i) acc[i] = 1.0f;
    v8f czero = {};

    for (int u2 = 0; u2 < UNITS / 2; ++u2) {
        // one b128 load covers units (2*u2) in .x/.y and (2*u2+1) in .z/.w
        float4 xv = xrow[u2];
        float4 pv = prow[u2];

        v2f A, B;
        // unit 2*u2
        A[0] = xv.x * W00 + xv.y * W10;   // xW[b,u,0] (hi lanes -> 0)
        A[1] = xv.x * W01 + xv.y * W11;   // xW[b,u,1]
        B[0] = pv.x * m;                  // xp[p,u,0] (hi lanes -> 0)
        B[1] = pv.y * m;
        v8f d = __builtin_amdgcn_wmma_f32_16x16x4_f32(
            false, A, false, B, (short)0, czero, false, false);
#pragma unroll
        for (int i = 0; i < 8; ++i) acc[i] *= d[i];

        // unit 2*u2+1
        A[0] = xv.z * W00 + xv.w * W10;
        A[1] = xv.z * W01 + xv.w * W11;
        B[0] = pv.z * m;
        B[1] = pv.w * m;
        d = __builtin_amdgcn_wmma_f32_16x16x4_f32(
            false, A, false, B, (short)0, czero, false, false);
#pragma unroll
        for (int i = 0; i < 8; ++i) acc[i] *= d[i];
    }

    // D layout: VGPR i, lanes 0-15 -> (M=i, N=lane); lanes 16-31 -> (M=8+i, N=lane-16)
    const int p  = pbase + l15;
    const int hi = lane >> 4;
    const float th = Th[p];
#pragma unroll
    for (int i = 0; i < 8; ++i) {
        int b = bbase + i + hi * 8;
        out[(size_t)b * NP + p] = acc[i] * th;
    }
}

// ---------------------------------------------------------------------------
extern "C" void kernel_launch(void* const* d_in, const int* in_sizes, int n_in,
                              void* d_out, int out_size, void* d_ws, size_t ws_size,
                              hipStream_t stream) {
    const float* x  = (const float*)d_in[0]; // (NX, UNITS, 2)
    const float* xp = (const float*)d_in[1]; // (NP, UNITS, 2)
    const float* W  = (const float*)d_in[2]; // (2, 2)
    float* out = (float*)d_out;              // (NX, NP)
    float* Th  = (float*)d_ws;               // NP floats of scratch

    statmech_th_kernel<<<1, 256, 0, stream>>>(xp, W, Th);
    // 256 output tiles of 16x16 -> 256 waves -> 32 blocks x 256 threads (8 wave32s)
    statmech_tv_kernel<<<32, 256, 0, stream>>>(x, xp, W, Th, out);
}